// GraphTransformerModel_82532091560101
// MI455X (gfx1250) — compile-verified
//
#include <hip/hip_runtime.h>
#include <hip/hip_bf16.h>

// ---------------------------------------------------------------------------
// GraphTransformerConv (PyG TransformerConv, concat, root_weight) + log_softmax
// N=100000 nodes, F=64, H=2 heads, C=5, E=3.2M edges.
//
// Per-node packed layout in scratch (48 floats, 192B rows, all 16B aligned):
//   [0..9]=q  [12..21]=k  [24..33]=v  [36..45]=acc (init = x@Wskip+bskip)
// Fused QKV+skip GEMM via V_WMMA_F32_16X16X4_F32 (exact fp32 matrix pipe).
// W is pre-packed in WMMA B-fragment order so each B operand is ONE
// ds_load_b64 into an aligned VGPR pair (no shuffle movs).
// Edge phases recompute the logit each pass; all hot data (~47MB) is
// L2-resident (192MB), so repeat passes run at L2/atomic rates.
// ---------------------------------------------------------------------------

typedef __attribute__((ext_vector_type(2))) float v2f;
typedef __attribute__((ext_vector_type(8))) float v8f;

#define HC 10      // H*C
#define PAD 48     // padded per-node row
#define RSQRT_C 0.44721359549995793f

// ---------------- small helpers ----------------
__device__ __forceinline__ unsigned fenc(float f) {
    unsigned u = __float_as_uint(f);
    return (u & 0x80000000u) ? ~u : (u | 0x80000000u);
}
__device__ __forceinline__ float fdec(unsigned e) {
    unsigned u = (e & 0x80000000u) ? (e & 0x7fffffffu) : ~e;
    return __uint_as_float(u);
}
__device__ __forceinline__ void load10(const float* __restrict__ p, float* r) {
    float4 a = *reinterpret_cast<const float4*>(p);
    float4 b = *reinterpret_cast<const float4*>(p + 4);
    float2 c = *reinterpret_cast<const float2*>(p + 8);
    r[0] = a.x; r[1] = a.y; r[2] = a.z; r[3] = a.w;
    r[4] = b.x; r[5] = b.y; r[6] = b.z; r[7] = b.w;
    r[8] = c.x; r[9] = c.y;
}

// logical padded weight element W[k][j], k in [0,64), j in [0,48)
__device__ __forceinline__ float wval(const float* __restrict__ Wq, const float* __restrict__ Wk,
                                      const float* __restrict__ Wv, const float* __restrict__ Ws,
                                      int k, int j) {
    if (j < 10)            return Wq[k * HC + j];
    if (j >= 12 && j < 22) return Wk[k * HC + (j - 12)];
    if (j >= 24 && j < 34) return Wv[k * HC + (j - 24)];
    if (j >= 36 && j < 46) return Ws[k * HC + (j - 36)];
    return 0.f;
}

// ---------------- pack W into WMMA B-fragment pair order + bias[48] ---------
// Wp2[(s*2 + half)*48 + n] = { W[4s + 2*half][n], W[4s + 2*half + 1][n] }
__global__ void pack_w_kernel(const float* __restrict__ Wq, const float* __restrict__ bq,
                              const float* __restrict__ Wk, const float* __restrict__ bk,
                              const float* __restrict__ Wv, const float* __restrict__ bv,
                              const float* __restrict__ Ws, const float* __restrict__ bs,
                              float2* __restrict__ Wp2, float* __restrict__ bp) {
    int i = blockIdx.x * blockDim.x + threadIdx.x;
    if (i < 16 * 2 * PAD) {
        int n    = i % PAD;
        int half = (i / PAD) & 1;
        int s    = i / (2 * PAD);
        int k    = 4 * s + 2 * half;
        float2 p;
        p.x = wval(Wq, Wk, Wv, Ws, k,     n);
        p.y = wval(Wq, Wk, Wv, Ws, k + 1, n);
        Wp2[i] = p;
    }
    if (i < PAD) {
        float v = 0.f;
        if (i < 10)                 v = bq[i];
        else if (i >= 12 && i < 22) v = bk[i - 12];
        else if (i >= 24 && i < 34) v = bv[i - 24];
        else if (i >= 36 && i < 46) v = bs[i - 36];
        bp[i] = v;
    }
}

// ---------------- init segment-max (encoded) and denominators ----------------
__global__ void init_stats_kernel(unsigned* __restrict__ amax, float* __restrict__ den, int n2) {
    int i = blockIdx.x * blockDim.x + threadIdx.x;
    if (i < n2) { amax[i] = 0u; den[i] = 0.f; }   // 0u encodes "below every float"
}

// ---------------- fused QKV+skip GEMM via V_WMMA_F32_16X16X4_F32 ----------------
__global__ __launch_bounds__(256) void gemm_qkvs_kernel(const float* __restrict__ x,
                                                        const float2* __restrict__ Wp2,
                                                        const float* __restrict__ bp,
                                                        float* __restrict__ QKVS,
                                                        int N, int NT) {
    __shared__ v2f  lwp[16 * 2 * PAD];   // B fragments in pair order (12 KB)
    __shared__ float lbias[PAD];
    for (int i = threadIdx.x; i < 16 * 2 * PAD; i += 256) {
        float2 p = Wp2[i];
        v2f q; q[0] = p.x; q[1] = p.y;
        lwp[i] = q;
    }
    if (threadIdx.x < PAD) lbias[threadIdx.x] = bp[threadIdx.x];
    __syncthreads();

    const int wave = threadIdx.x >> 5;
    const int lane = threadIdx.x & 31;
    const int rt = blockIdx.x * 8 + wave;
    if (rt >= NT) return;                 // whole-wave exit: EXEC all-ones inside

    const int row0 = rt * 16;
    const int half = lane >> 4;           // 0: lanes 0-15, 1: lanes 16-31
    const int m    = lane & 15;
    const int klo  = half * 2;

    // A fragments: lane holds {x[row0+m][4s+klo], x[row0+m][4s+klo+1]}
    int arow = row0 + m;
    if (arow >= N) arow = N - 1;          // clamp (stores are guarded)
    const float* xr = x + (size_t)arow * 64 + klo;
    v2f a[16];
#pragma unroll
    for (int s = 0; s < 16; ++s)
        a[s] = *reinterpret_cast<const v2f*>(xr + 4 * s);

    // bias-initialized accumulators (bias depends only on output column)
    v8f acc[3];
#pragma unroll
    for (int t = 0; t < 3; ++t) {
        float bvv = lbias[16 * t + m];
#pragma unroll
        for (int r = 0; r < 8; ++r) acc[t][r] = bvv;
    }

    const v2f* bfrag = lwp + (size_t)half * PAD + m;   // + s*(2*PAD) + 16*t
#pragma unroll
    for (int s = 0; s < 16; ++s) {
#pragma unroll
        for (int t = 0; t < 3; ++t) {
            v2f b = bfrag[s * (2 * PAD) + 16 * t];     // one ds_load_b64
            acc[t] = __builtin_amdgcn_wmma_f32_16x16x4_f32(
                false, a[s], false, b, (short)0, acc[t], false, false);
        }
    }

    // D layout: VGPR r -> row = row0 + half*8 + r, col = 16*t + m
#pragma unroll
    for (int t = 0; t < 3; ++t) {
#pragma unroll
        for (int r = 0; r < 8; ++r) {
            int row = row0 + half * 8 + r;
            if (row < N) QKVS[(size_t)row * PAD + 16 * t + m] = acc[t][r];
        }
    }
}

// ---------------- edge pass 1: segment max of attention logits ----------------
__global__ void edge_max_kernel(const int* __restrict__ ei, const float* __restrict__ QKVS,
                                unsigned* __restrict__ amax, int E) {
    int e = blockIdx.x * blockDim.x + threadIdx.x;
    if (e >= E) return;
    int s = ei[e], d = ei[E + e];
    float q[10], k[10];
    load10(QKVS + (size_t)d * PAD, q);
    load10(QKVS + (size_t)s * PAD + 12, k);
#pragma unroll
    for (int h = 0; h < 2; ++h) {
        float al = 0.f;
#pragma unroll
        for (int c = 0; c < 5; ++c) al += q[h * 5 + c] * k[h * 5 + c];
        al *= RSQRT_C;
        atomicMax(&amax[d * 2 + h], fenc(al));
    }
}

// ---------------- edge pass 2: segment sum of exp(alpha - amax) ----------------
__global__ void edge_den_kernel(const int* __restrict__ ei, const float* __restrict__ QKVS,
                                const unsigned* __restrict__ amax, float* __restrict__ den, int E) {
    int e = blockIdx.x * blockDim.x + threadIdx.x;
    if (e >= E) return;
    int s = ei[e], d = ei[E + e];
    float q[10], k[10];
    load10(QKVS + (size_t)d * PAD, q);
    load10(QKVS + (size_t)s * PAD + 12, k);
#pragma unroll
    for (int h = 0; h < 2; ++h) {
        float al = 0.f;
#pragma unroll
        for (int c = 0; c < 5; ++c) al += q[h * 5 + c] * k[h * 5 + c];
        al *= RSQRT_C;
        float ex = expf(al - fdec(amax[d * 2 + h]));
        unsafeAtomicAdd(&den[d * 2 + h], ex);
    }
}

// ---------------- edge pass 3: weighted message scatter-add ----------------
__global__ void edge_scatter_kernel(const int* __restrict__ ei, float* __restrict__ QKVS,
                                    const unsigned* __restrict__ amax,
                                    const float* __restrict__ den, int E) {
    int e = blockIdx.x * blockDim.x + threadIdx.x;
    if (e >= E) return;
    int s = ei[e], d = ei[E + e];
    float q[10], k[10], v[10];
    load10(QKVS + (size_t)d * PAD, q);
    load10(QKVS + (size_t)s * PAD + 12, k);
    load10(QKVS + (size_t)s * PAD + 24, v);
    float* accp = QKVS + (size_t)d * PAD + 36;
#pragma unroll
    for (int h = 0; h < 2; ++h) {
        float al = 0.f;
#pragma unroll
        for (int c = 0; c < 5; ++c) al += q[h * 5 + c] * k[h * 5 + c];
        al *= RSQRT_C;
        float attn = expf(al - fdec(amax[d * 2 + h])) / (den[d * 2 + h] + 1e-16f);
#pragma unroll
        for (int c = 0; c < 5; ++c)
            unsafeAtomicAdd(&accp[h * 5 + c], v[h * 5 + c] * attn);
    }
}

// ---------------- final: log_softmax over the 10 channels ----------------
__global__ void logsoftmax_kernel(const float* __restrict__ QKVS, float* __restrict__ out, int N) {
    int n = blockIdx.x * blockDim.x + threadIdx.x;
    if (n >= N) return;
    float a[10];
    load10(QKVS + (size_t)n * PAD + 36, a);
    float mx = a[0];
#pragma unroll
    for (int j = 1; j < 10; ++j) mx = fmaxf(mx, a[j]);
    float sum = 0.f;
#pragma unroll
    for (int j = 0; j < 10; ++j) sum += expf(a[j] - mx);
    float ls = logf(sum);
#pragma unroll
    for (int j = 0; j < 10; ++j) out[(size_t)n * 10 + j] = a[j] - mx - ls;
}

// ---------------------------------------------------------------------------
extern "C" void kernel_launch(void* const* d_in, const int* in_sizes, int n_in,
                              void* d_out, int out_size, void* d_ws, size_t ws_size,
                              hipStream_t stream) {
    const float* x  = (const float*)d_in[0];
    const int*   ei = (const int*)d_in[1];
    const float* Wq = (const float*)d_in[2];
    const float* bq = (const float*)d_in[3];
    const float* Wk = (const float*)d_in[4];
    const float* bk = (const float*)d_in[5];
    const float* Wv = (const float*)d_in[6];
    const float* bv = (const float*)d_in[7];
    const float* Ws = (const float*)d_in[8];
    const float* bs = (const float*)d_in[9];
    float* out = (float*)d_out;

    const int N = in_sizes[0] / 64;   // 100000
    const int E = in_sizes[1] / 2;    // 3200000

    // workspace carve-out (~21 MB)
    char* ws = (char*)d_ws;
    float2*   Wp2  = (float2*)ws;                            // 16*2*48 float2 (12 KB)
    float*    bp   = (float*)(ws + 13 * 1024);               // 48 floats
    float*    QKVS = (float*)(ws + 16 * 1024);               // N*48 floats
    unsigned* AMAX = (unsigned*)(ws + 16 * 1024 + (size_t)N * PAD * sizeof(float)); // N*2
    float*    DEN  = (float*)((char*)AMAX + (size_t)N * 2 * sizeof(unsigned));      // N*2

    pack_w_kernel<<<(16 * 2 * PAD + 255) / 256, 256, 0, stream>>>(Wq, bq, Wk, bk, Wv, bv, Ws, bs, Wp2, bp);
    init_stats_kernel<<<(2 * N + 255) / 256, 256, 0, stream>>>(AMAX, DEN, 2 * N);

    const int NT = (N + 15) / 16;
    gemm_qkvs_kernel<<<(NT + 7) / 8, 256, 0, stream>>>(x, Wp2, bp, QKVS, N, NT);

    const int eb = (E + 255) / 256;
    edge_max_kernel<<<eb, 256, 0, stream>>>(ei, QKVS, AMAX, E);
    edge_den_kernel<<<eb, 256, 0, stream>>>(ei, QKVS, AMAX, DEN, E);
    edge_scatter_kernel<<<eb, 256, 0, stream>>>(ei, QKVS, AMAX, DEN, E);

    logsoftmax_kernel<<<(N + 255) / 256, 256, 0, stream>>>(QKVS, out, N);
}